// TCFPEmbedding_18949395710715
// MI455X (gfx1250) — compile-verified
//
#include <hip/hip_runtime.h>
#include <cstdint>

// ---------------------------------------------------------------------------
// TCFP-12 fake-quantized embedding gather for MI455X (gfx1250, wave32).
//
// Bandwidth-bound: ~100 MB gathered reads (weight table, 154 MB, is kept
// L2-resident) + ~100 MB streamed output (non-temporal stores so it does not
// evict the table from the 192 MB L2). Each workgroup handles TOK_PER_WG
// tokens with double-buffered CDNA5 async global->LDS copies
// (global_load_async_to_lds_b64 / s_wait_asynccnt): the copy for token i+1 is
// in flight while token i is quantized.
//
// blockDim = emb/2 (=384 for emb=768) -> 12 wave32 waves; wave w quantizes
// 64-element block w of the row, lane l owns elements 2l, 2l+1, and each lane
// async-copies exactly the 8 bytes it later consumes (no inter-wave LDS
// dependencies in the steady-state loop).
// ---------------------------------------------------------------------------

typedef float v2f __attribute__((ext_vector_type(2)));

constexpr int TOK_PER_WG = 8;
constexpr int MAX_HALF = 1536;  // supports emb up to 3072 floats

// NF4 codebook (QLoRA normal-float-4 levels, sorted). constexpr copy folds
// the midpoint comparisons to literals; __device__ copy feeds the LDS table.
constexpr float kNF4[16] = {
    -1.0f, -0.6961928009986877f, -0.5250730514526367f, -0.39491748809814453f,
    -0.28444138169288635f, -0.18477343022823334f, -0.09105003625154495f, 0.0f,
    0.07958029955625534f, 0.16093020141124725f, 0.24611230194568634f,
    0.33791524171829224f, 0.44070982933044434f, 0.5626170039176941f,
    0.8481764793395996f, 1.0f};

__device__ const float g_nf4_levels[16] = {
    -1.0f, -0.6961928009986877f, -0.5250730514526367f, -0.39491748809814453f,
    -0.28444138169288635f, -0.18477343022823334f, -0.09105003625154495f, 0.0f,
    0.07958029955625534f, 0.16093020141124725f, 0.24611230194568634f,
    0.33791524171829224f, 0.44070982933044434f, 0.5626170039176941f,
    0.8481764793395996f, 1.0f};

__device__ __forceinline__ float wave_reduce_sum(float x) {
#pragma unroll
  for (int m = 16; m >= 1; m >>= 1) x += __shfl_xor(x, m, 32);
  return x;
}

__device__ __forceinline__ float wave_reduce_max(float x) {
#pragma unroll
  for (int m = 16; m >= 1; m >>= 1) x = fmaxf(x, __shfl_xor(x, m, 32));
  return x;
}

// Issue this lane's 8-byte slice of row `indices[token]` into LDS slot.
// ASYNCcnt += 1; completion is in-order with other async loads of this wave.
__device__ __forceinline__ void issue_row_async(const int* __restrict__ indices,
                                                const float* __restrict__ weight,
                                                int token, int num_tokens,
                                                int emb, int t,
                                                const float* slot) {
  const int tk = (token < num_tokens) ? token : (num_tokens - 1);  // clamp
  const int row = indices[tk];  // uniform per workgroup
  const uint64_t gaddr = (uint64_t)(uintptr_t)weight +
                         ((uint64_t)row * (uint64_t)emb + (uint64_t)(2 * t)) *
                             4ull;
  const uint32_t laddr = (uint32_t)(uintptr_t)slot + (uint32_t)t * 8u;
  asm volatile("global_load_async_to_lds_b64 %0, %1, off"
               :
               : "v"(laddr), "v"(gaddr)
               : "memory");
}

__global__ __launch_bounds__(512) void tcfp12_embed_kernel(
    const int* __restrict__ indices, const float* __restrict__ weight,
    float* __restrict__ out, int emb, int num_tokens) {
  __shared__ float lvl[16];
  __shared__ __align__(16) float buf[2][MAX_HALF * 2];

  const int t = threadIdx.x;        // 0 .. emb/2 - 1
  const int half = blockDim.x;      // emb/2

  if (t < 16) lvl[t] = g_nf4_levels[t];

  const int tok0 = blockIdx.x * TOK_PER_WG;
  if (tok0 >= num_tokens) return;

  // Prime slot 0 for the first token; the barrier for the level table
  // overlaps with the async copy latency.
  issue_row_async(indices, weight, tok0, num_tokens, emb, t, &buf[0][0]);
  __syncthreads();

#pragma unroll
  for (int i = 0; i < TOK_PER_WG; ++i) {
    const int token = tok0 + i;
    if (token >= num_tokens) break;

    const bool has_next = (i + 1 < TOK_PER_WG) && (token + 1 < num_tokens);
    if (has_next) {
      // Prefetch next row into the other slot while we process this one.
      issue_row_async(indices, weight, token + 1, num_tokens, emb, t,
                      &buf[(i + 1) & 1][0]);
      // Allow the just-issued copy to remain in flight; the older one (our
      // current slot) must have completed. "memory" clobber also fences the
      // subsequent ds_load against compiler reordering.
      asm volatile("s_wait_asynccnt 0x1" ::: "memory");
    } else {
      asm volatile("s_wait_asynccnt 0x0" ::: "memory");
    }

    const float2 v = reinterpret_cast<const float2*>(&buf[i & 1][0])[t];
    const float a = v.x;
    const float b = v.y;

    // ---- per-64-block statistics (one wave == one block, 2 elems/lane) ----
    const float s = wave_reduce_sum(a + b);
    const float ss = wave_reduce_sum(a * a + b * b);
    const float am = wave_reduce_max(fmaxf(fabsf(a), fabsf(b)));

    const float mean = s * 0.015625f;                             // /64
    const float var = fmaxf(ss * 0.015625f - mean * mean, 0.0f);  // pop. var
    float scale = fminf(am, 3.0f * sqrtf(var));                   // sigma = 3
    scale = fmaxf(scale, 1e-12f);

    // ---- int8 symmetric main component (RTE round == jnp.round) ----
    const float inv = 127.0f / scale;
    const float stp = scale * (1.0f / 127.0f);
    const float qa = fminf(fmaxf(rintf(a * inv), -127.0f), 127.0f);
    const float qb = fminf(fmaxf(rintf(b * inv), -127.0f), 127.0f);
    const float ma = qa * stp;
    const float mb = qb * stp;

    // ---- NF4 residual ----
    const float ra = a - ma;
    const float rb = b - mb;
    float rmax = wave_reduce_max(fmaxf(fabsf(ra), fabsf(rb)));
    rmax = fmaxf(rmax, 1e-12f);
    const float na = ra / rmax;
    const float nb = rb / rmax;

    // searchsorted(midpoints, rn, side='left') == # midpoints strictly < rn
    int ia = 0, ib = 0;
#pragma unroll
    for (int j = 0; j < 15; ++j) {
      const float mid = 0.5f * (kNF4[j] + kNF4[j + 1]);  // folded literal
      ia += (na > mid) ? 1 : 0;
      ib += (nb > mid) ? 1 : 0;
    }

    const float da = fmaf(lvl[ia], rmax, ma);
    const float db = fmaf(lvl[ib], rmax, mb);

    // Non-temporal store: keep the 100 MB output stream out of L2 so the
    // 154 MB embedding table stays L2-resident for the gather.
    v2f o;
    o.x = da;
    o.y = db;
    v2f* dst = reinterpret_cast<v2f*>(out) + (size_t)token * (size_t)half + t;
    __builtin_nontemporal_store(o, dst);
  }
}

extern "C" void kernel_launch(void* const* d_in, const int* in_sizes, int n_in,
                              void* d_out, int out_size, void* d_ws,
                              size_t ws_size, hipStream_t stream) {
  (void)n_in;
  (void)d_ws;
  (void)ws_size;
  const int* indices = (const int*)d_in[0];     // int32 [batch*seq]
  const float* weight = (const float*)d_in[1];  // fp32 [num_embeddings, emb]
  float* out = (float*)d_out;                   // fp32 [batch*seq, emb]

  const int num_tokens = in_sizes[0];
  if (num_tokens <= 0) return;
  const int emb = out_size / num_tokens;  // 768 for this harness
  const int threads = emb / 2;            // 384 -> 12 wave32 waves
  if (threads <= 0 || threads > 512) return;

  const int blocks = (num_tokens + TOK_PER_WG - 1) / TOK_PER_WG;
  tcfp12_embed_kernel<<<blocks, threads, 0, stream>>>(indices, weight, out,
                                                      emb, num_tokens);
}